// LayerNorm_28260884808104
// MI455X (gfx1250) — compile-verified
//
#include <hip/hip_runtime.h>
#include <cstdint>

// Segment LayerNorm over ragged CSR segments (N=2M rows, S=8192, D=64, f32).
// Memory-bound: floor = 1.07 GB HBM -> ~46us @ 23.3 TB/s on MI455X.
// CDNA5 paths: global_load_async_to_lds_b128 (ASYNCcnt) stages each segment
// into LDS once; the fused stats sweep and the normalize sweep then run out
// of LDS, keeping HBM traffic at the roofline floor. The instruction offset
// field is applied to BOTH the LDS and global addresses (ISA 08 §4.4), so we
// issue 4 DMA chunks per address pair. 96 KB staging -> 3 blocks/WGP (314 KB
// of the 320 KB pool) so the asynccnt wait of one block overlaps the compute
// of two others. No GEMM -> no WMMA (it cannot reduce bytes moved).

#define DFEAT     64
#define NTHREADS  256
#define LDS_ROWS  384          // 384 rows * 256B = 96 KB staging
#define NGRP      16           // row-groups in the stats sweep (t >> 4)
#define EPS       1e-5f

__device__ __forceinline__ unsigned lds_byte_off(const void* p) {
  // Flat LDS addresses are {SHARED_BASE[63:32], lds_offset[31:0]} (ISA 10.2):
  // truncating a generic pointer into __shared__ yields the wave-relative LDS
  // byte address the async-DMA instruction expects in VDST.
  return (unsigned)(uintptr_t)p;
}

__global__ __launch_bounds__(NTHREADS, 1)
void seg_layernorm_kernel(const float* __restrict__ x,
                          const int*   __restrict__ offsets,
                          const float* __restrict__ weight,
                          const float* __restrict__ bias,
                          float*       __restrict__ out)
{
  __shared__ __align__(16) float smem[LDS_ROWS * DFEAT];   // 96 KB staging
  __shared__ __align__(16) float red_s[NGRP * DFEAT];      // 4 KB
  __shared__ __align__(16) float red_q[NGRP * DFEAT];      // 4 KB
  __shared__ __align__(16) float s_scale[DFEAT];
  __shared__ __align__(16) float s_shift[DFEAT];

  const int seg   = blockIdx.x;
  const int t     = threadIdx.x;
  const int begin = (seg == 0) ? 0 : offsets[seg - 1];
  const int end   = offsets[seg];
  const int len   = end - begin;
  const int nch   = len * (DFEAT / 4);     // float4 chunks in this segment
  const bool staged = (len <= LDS_ROWS);

  const float4* __restrict__ xin4 =
      reinterpret_cast<const float4*>(x) + (size_t)begin * (DFEAT / 4);
  float4* __restrict__ out4 =
      reinterpret_cast<float4*>(out) + (size_t)begin * (DFEAT / 4);

  // ---- Stage segment into LDS with CDNA5 async global->LDS DMA -----------
  if (staged) {
    const uint64_t srcbase = (uint64_t)(uintptr_t)xin4;
    int i = t;
    // 4 issues share one (lds,voff) pair; offset applies to LDS and global.
    for (; i + 3 * NTHREADS < nch; i += 4 * NTHREADS) {
      unsigned ldsa = lds_byte_off(&smem[i * 4]);
      unsigned voff = (unsigned)(i * 16);
      asm volatile(
          "global_load_async_to_lds_b128 %0, %1, %2 offset:0\n\t"
          "global_load_async_to_lds_b128 %0, %1, %2 offset:4096\n\t"
          "global_load_async_to_lds_b128 %0, %1, %2 offset:8192\n\t"
          "global_load_async_to_lds_b128 %0, %1, %2 offset:12288"
          :: "v"(ldsa), "v"(voff), "s"(srcbase) : "memory");
    }
    for (; i < nch; i += NTHREADS) {
      unsigned ldsa = lds_byte_off(&smem[i * 4]);
      unsigned voff = (unsigned)(i * 16);
      asm volatile("global_load_async_to_lds_b128 %0, %1, %2 offset:0"
                   :: "v"(ldsa), "v"(voff), "s"(srcbase) : "memory");
    }
    asm volatile("s_wait_asynccnt 0" ::: "memory");
  }
  __syncthreads();

  // ---- Fused stats sweep: per-feature sum(x) and sum(x^2), float4-wide ---
  // Thread owns features [4*c4, 4*c4+3], rows g, g+16, g+32, ...
  const int c4 = t & (NGRP - 1);   // float4 chunk within a row (invariant)
  const int g  = t >> 4;           // row group 0..15
  {
    float sx = 0.f, sy = 0.f, sz = 0.f, sw = 0.f;
    float qx = 0.f, qy = 0.f, qz = 0.f, qw = 0.f;
    if (staged) {
      for (int r = g; r < len; r += NGRP) {
        float4 v = *reinterpret_cast<const float4*>(&smem[r * DFEAT + c4 * 4]);
        sx += v.x; sy += v.y; sz += v.z; sw += v.w;
        qx = fmaf(v.x, v.x, qx); qy = fmaf(v.y, v.y, qy);
        qz = fmaf(v.z, v.z, qz); qw = fmaf(v.w, v.w, qw);
      }
    } else {
      for (int r = g; r < len; r += NGRP) {
        float4 v = xin4[r * (DFEAT / 4) + c4];
        sx += v.x; sy += v.y; sz += v.z; sw += v.w;
        qx = fmaf(v.x, v.x, qx); qy = fmaf(v.y, v.y, qy);
        qz = fmaf(v.z, v.z, qz); qw = fmaf(v.w, v.w, qw);
      }
    }
    float4 s4; s4.x = sx; s4.y = sy; s4.z = sz; s4.w = sw;
    float4 q4; q4.x = qx; q4.y = qy; q4.z = qz; q4.w = qw;
    *reinterpret_cast<float4*>(&red_s[g * DFEAT + c4 * 4]) = s4;
    *reinterpret_cast<float4*>(&red_q[g * DFEAT + c4 * 4]) = q4;
  }
  __syncthreads();

  // ---- Reduce 16 row-groups; build scale/shift ---------------------------
  if (t < DFEAT) {
    float s0 = 0.f, q0 = 0.f;
#pragma unroll
    for (int k = 0; k < NGRP; ++k) {
      s0 += red_s[k * DFEAT + t];
      q0 += red_q[k * DFEAT + t];
    }
    float inv  = 1.0f / (float)len;
    float mean = s0 * inv;
    float var  = fmaxf(q0 * inv - mean * mean, 0.0f);
    float rstd = 1.0f / sqrtf(var + EPS);
    float sc   = rstd * weight[t];
    s_scale[t] = sc;
    s_shift[t] = bias[t] - mean * sc;
  }
  __syncthreads();

  // ---- Normalize sweep: out = x*scale + shift, float4 coalesced ----------
  // i & 15 == t & 15 for stride-256 iteration -> scale/shift are invariant:
  // load them once as float4 (one ds_load_b128 each).
  const float4 sc4 = *reinterpret_cast<const float4*>(&s_scale[c4 * 4]);
  const float4 sh4 = *reinterpret_cast<const float4*>(&s_shift[c4 * 4]);
  if (staged) {
    for (int i = t; i < nch; i += NTHREADS) {
      float4 v = *reinterpret_cast<const float4*>(&smem[i * 4]);
      float4 o;
      o.x = fmaf(v.x, sc4.x, sh4.x);
      o.y = fmaf(v.y, sc4.y, sh4.y);
      o.z = fmaf(v.z, sc4.z, sh4.z);
      o.w = fmaf(v.w, sc4.w, sh4.w);
      out4[i] = o;
    }
  } else {
    for (int i = t; i < nch; i += NTHREADS) {
      float4 v = xin4[i];
      float4 o;
      o.x = fmaf(v.x, sc4.x, sh4.x);
      o.y = fmaf(v.y, sc4.y, sh4.y);
      o.z = fmaf(v.z, sc4.z, sh4.z);
      o.w = fmaf(v.w, sc4.w, sh4.w);
      out4[i] = o;
    }
  }
}

extern "C" void kernel_launch(void* const* d_in, const int* in_sizes, int n_in,
                              void* d_out, int out_size, void* d_ws, size_t ws_size,
                              hipStream_t stream) {
  const float* x       = (const float*)d_in[0];
  const int*   offsets = (const int*)d_in[1];
  const float* weight  = (const float*)d_in[2];
  const float* bias    = (const float*)d_in[3];
  float*       out     = (float*)d_out;
  const int S = in_sizes[1];   // number of segments (offsets length)
  seg_layernorm_kernel<<<S, NTHREADS, 0, stream>>>(x, offsets, weight, bias, out);
}